// Structure2Vec_45191645888914
// MI455X (gfx1250) — compile-verified
//
#include <hip/hip_runtime.h>
#include <hip/hip_bf16.h>

typedef __attribute__((ext_vector_type(2))) float v2f;
typedef __attribute__((ext_vector_type(8))) float v8f;

#define FDIM 128
#define SLOPE 0.01f
#define LDSA 132            // padded row stride (floats) for A tile in LDS
#define LDSW 132            // padded col stride (floats) for transposed W in LDS
#define ROWS_PER_BLOCK 64   // 4 row-groups x 16 rows

__device__ __forceinline__ float lrelu(float v) { return v >= 0.0f ? v : SLOPE * v; }

__global__ __launch_bounds__(256) void fill_zero_kernel(float* __restrict__ p, int n) {
    int i = blockIdx.x * blockDim.x + threadIdx.x;
    if (i < n) p[i] = 0.0f;
}

__global__ __launch_bounds__(256) void degree_kernel(const int* __restrict__ edst,
                                                     float* __restrict__ deg, int nE) {
    int e = blockIdx.x * blockDim.x + threadIdx.x;
    if (e < nE) unsafeAtomicAdd(&deg[edst[e]], 1.0f);
}

__global__ __launch_bounds__(256) void inv_kernel(float* __restrict__ inv, int n) {
    int i = blockIdx.x * blockDim.x + threadIdx.x;
    if (i < n) {
        float d = inv[i];
        inv[i] = d > 0.0f ? 1.0f / d : 0.0f;
    }
}

// One wave (32 lanes) per edge; each lane handles 4 consecutive features.
// h rows are 512B, L2-resident (51MB << 192MB L2), so atomics land in L2.
__global__ __launch_bounds__(256) void aggregate_kernel(const float* __restrict__ h,
                                                        const int* __restrict__ esrc,
                                                        const int* __restrict__ edst,
                                                        float* __restrict__ s, int nE) {
    int gid  = blockIdx.x * blockDim.x + threadIdx.x;
    int edge = gid >> 5;
    int lane = gid & 31;
    if (edge >= nE) return;
    int srow = esrc[edge];
    int drow = edst[edge];
    const float4 hv = ((const float4*)(h + (size_t)srow * FDIM))[lane];
    float* sp = s + (size_t)drow * FDIM + lane * 4;
    unsafeAtomicAdd(sp + 0, hv.x);
    unsafeAtomicAdd(sp + 1, hv.y);
    unsafeAtomicAdd(sp + 2, hv.z);
    unsafeAtomicAdd(sp + 3, hv.w);
}

// mode 0:  H = lrelu(A @ W + b)                 (fc1)
// mode 1:  H = lrelu((A * inv_row) @ W + b + H) (message-passing update, in place)
// Full fp32 via V_WMMA_F32_16X16X4_F32. W is staged TRANSPOSED in LDS so both
// A and B fragments are contiguous 8B-aligned pairs -> single ds_load_b64 each,
// no register shuffling in front of the WMMAs.
__global__ __launch_bounds__(256) void gemm_update_kernel(
        const float* __restrict__ A, const float* __restrict__ W,
        const float* __restrict__ bias, const float* __restrict__ inv,
        float* __restrict__ Hbuf, int n, int mode) {
    extern __shared__ float smem[];
    float* As = smem;                          // [64][LDSA]   row-major A tile
    float* Wt = smem + ROWS_PER_BLOCK * LDSA;  // [128][LDSW]  Wt[col][k] = W[k][col]

    const int tid      = threadIdx.x;
    const int rowBlock = blockIdx.x * ROWS_PER_BLOCK;

    // Stage W transposed: idx enumerates (colGroup, row) so that within a wave
    // lanes walk consecutive rows -> conflict-free LDS stores (banks r+const).
    #pragma unroll
    for (int i = 0; i < 16; ++i) {
        int idx = i * 256 + tid;           // 0..4095
        int c4 = idx >> 7;                 // 0..31 : 4-column group
        int r  = idx & 127;                // 0..127: W row (= k)
        float4 w4 = ((const float4*)(W + (size_t)r * FDIM))[c4];
        Wt[(c4 * 4 + 0) * LDSW + r] = w4.x;
        Wt[(c4 * 4 + 1) * LDSW + r] = w4.y;
        Wt[(c4 * 4 + 2) * LDSW + r] = w4.z;
        Wt[(c4 * 4 + 3) * LDSW + r] = w4.w;
    }
    // Stage A tile row-major (coalesced float4); scale by inv in mode 1.
    #pragma unroll
    for (int i = 0; i < 8; ++i) {
        int idx = i * 256 + tid;
        int r = idx >> 5, c4 = idx & 31;
        int grow = rowBlock + r;
        float4 a4 = make_float4(0.0f, 0.0f, 0.0f, 0.0f);
        if (grow < n) {
            a4 = ((const float4*)(A + (size_t)grow * FDIM))[c4];
            if (mode) {
                float sc = inv[grow];
                a4.x *= sc; a4.y *= sc; a4.z *= sc; a4.w *= sc;
            }
        }
        *(float4*)&As[r * LDSA + c4 * 4] = a4;
    }
    __syncthreads();

    const int wave     = tid >> 5;
    const int lane     = tid & 31;
    const int half     = lane >> 4;   // lanes 0-15: K=+0/+1, lanes 16-31: K=+2/+3
    const int lr       = lane & 15;
    const int rowGroup = wave >> 1;   // 0..3
    const int colHalf  = wave & 1;    // 0..1
    const int colBase  = colHalf * 64;

    v8f acc[4] = {};
    const float* Arow = &As[(rowGroup * 16 + lr) * LDSA];
    const float* Wc0  = &Wt[(colBase + 0 * 16 + lr) * LDSW];
    const float* Wc1  = &Wt[(colBase + 1 * 16 + lr) * LDSW];
    const float* Wc2  = &Wt[(colBase + 2 * 16 + lr) * LDSW];
    const float* Wc3  = &Wt[(colBase + 3 * 16 + lr) * LDSW];

    #pragma unroll
    for (int k = 0; k < FDIM; k += 4) {
        const int kk = k + 2 * half;                  // even -> 8B aligned
        v2f a  = *(const v2f*)(Arow + kk);
        v2f b0 = *(const v2f*)(Wc0 + kk);
        v2f b1 = *(const v2f*)(Wc1 + kk);
        v2f b2 = *(const v2f*)(Wc2 + kk);
        v2f b3 = *(const v2f*)(Wc3 + kk);
        acc[0] = __builtin_amdgcn_wmma_f32_16x16x4_f32(false, a, false, b0, (short)0, acc[0], false, false);
        acc[1] = __builtin_amdgcn_wmma_f32_16x16x4_f32(false, a, false, b1, (short)0, acc[1], false, false);
        acc[2] = __builtin_amdgcn_wmma_f32_16x16x4_f32(false, a, false, b2, (short)0, acc[2], false, false);
        acc[3] = __builtin_amdgcn_wmma_f32_16x16x4_f32(false, a, false, b3, (short)0, acc[3], false, false);
    }

    // C/D layout: lane = (half*16 + lr): N = lr, M = half*8 + v for VGPR v.
    #pragma unroll
    for (int t = 0; t < 4; ++t) {
        const int col = colBase + t * 16 + lr;
        const float bv = bias[col];
        #pragma unroll
        for (int v = 0; v < 8; ++v) {
            const int grow = rowBlock + rowGroup * 16 + half * 8 + v;
            if (grow < n) {
                float val = acc[t][v] + bv;
                if (mode) val += Hbuf[(size_t)grow * FDIM + col];
                Hbuf[(size_t)grow * FDIM + col] = lrelu(val);
            }
        }
    }
}

extern "C" void kernel_launch(void* const* d_in, const int* in_sizes, int n_in,
                              void* d_out, int out_size, void* d_ws, size_t ws_size,
                              hipStream_t stream) {
    const float* x    = (const float*)d_in[0];
    const float* W1   = (const float*)d_in[1];
    const float* b1   = (const float*)d_in[2];
    const float* W2   = (const float*)d_in[3];
    const float* b2   = (const float*)d_in[4];
    const int*   esrc = (const int*)d_in[5];
    const int*   edst = (const int*)d_in[6];
    float*       h    = (float*)d_out;

    const int n  = in_sizes[0] / FDIM;  // 100000 nodes
    const int nE = in_sizes[5];         // 1600000 edges

    float* s   = (float*)d_ws;          // [n, FDIM] neighbor sums
    float* inv = s + (size_t)n * FDIM;  // [n] 1/deg

    const int smemBytes  = (ROWS_PER_BLOCK * LDSA + FDIM * LDSW) * (int)sizeof(float);
    const int gemmBlocks = (n + ROWS_PER_BLOCK - 1) / ROWS_PER_BLOCK;

    hipFuncSetAttribute((const void*)gemm_update_kernel,
                        hipFuncAttributeMaxDynamicSharedMemorySize, smemBytes);

    // degree -> 1/deg (reuse inv buffer for the counts)
    fill_zero_kernel<<<(n + 255) / 256, 256, 0, stream>>>(inv, n);
    degree_kernel<<<(nE + 255) / 256, 256, 0, stream>>>(edst, inv, nE);
    inv_kernel<<<(n + 255) / 256, 256, 0, stream>>>(inv, n);

    // h = lrelu(x @ W1 + b1)
    gemm_update_kernel<<<gemmBlocks, 256, smemBytes, stream>>>(x, W1, b1, inv, h, n, 0);

    const int sElems     = n * FDIM;
    const int aggThreads = nE * 32;
    for (int it = 0; it < 3; ++it) {
        fill_zero_kernel<<<(sElems + 255) / 256, 256, 0, stream>>>(s, sElems);
        aggregate_kernel<<<(aggThreads + 255) / 256, 256, 0, stream>>>(h, esrc, edst, s, nE);
        gemm_update_kernel<<<gemmBlocks, 256, smemBytes, stream>>>(s, W2, b2, inv, h, n, 1);
    }
}